// RecursiveByCorrelationImputation_73899207295513
// MI455X (gfx1250) — compile-verified
//
#include <hip/hip_runtime.h>
#include <hip/hip_bf16.h>

#define Dd 512
#define Kk 128
#define Bb 4096

typedef float v2f __attribute__((ext_vector_type(2)));
typedef float v8f __attribute__((ext_vector_type(8)));

// ---------------------------------------------------------------------------
// Kernel 1: sequential column selection (argmax over masked corr), plus
// gathering of: colseq, b[colseq], transposed triangular wseqT[t][s] =
// w[col_s,col_t] (s<t), and the masked N-major weight panel
// wmT[n][d] = w[d,col_n] * known0[d].  Single block; O(K*(D+K)).
// ---------------------------------------------------------------------------
__global__ void __launch_bounds__(256) seq_kernel(
    const float* __restrict__ in0, const float* __restrict__ w,
    const float* __restrict__ bvec, const float* __restrict__ corr,
    int* __restrict__ g_colseq, float* __restrict__ g_bsel,
    float* __restrict__ g_wseqT, float* __restrict__ g_wmT)
{
    __shared__ float s_miss[Dd];
    __shared__ float s_known0[Dd];
    __shared__ int   s_mlist[Kk];
    __shared__ float s_bestv[Kk];
    __shared__ int   s_bestr[Kk];
    __shared__ int   s_alive[Kk];
    __shared__ int   s_colseq[Kk];
    __shared__ int   s_col;

    const int tid = threadIdx.x;
    for (int d = tid; d < Dd; d += 256) {
        float mv = in0[Dd + d];            // inputs[0,1,d]
        s_miss[d]   = mv;
        s_known0[d] = (mv > 0.5f) ? 0.0f : 1.0f;
    }
    __syncthreads();
    if (tid == 0) {
        int cnt = 0;
        for (int d = 0; d < Dd; ++d)
            if (s_miss[d] > 0.5f) { if (cnt < Kk) s_mlist[cnt] = d; ++cnt; }
    }
    __syncthreads();
    // Initial best known-row per missing column (row-major-first tiebreak).
    if (tid < Kk) {
        const int j = s_mlist[tid];
        float bv = -3.0f; int br = Dd;
        for (int i = 0; i < Dd; ++i) {
            if (s_known0[i] > 0.5f) {
                float v = corr[i * Dd + j];
                if (v > bv) { bv = v; br = i; }
            }
        }
        s_bestv[tid] = bv; s_bestr[tid] = br; s_alive[tid] = 1;
    }
    __syncthreads();
    for (int t = 0; t < Kk; ++t) {
        if (tid == 0) {
            float bv = -4.0f; long long bidx = 0x7fffffffffLL; int bslot = 0;
            for (int sl = 0; sl < Kk; ++sl) {
                if (s_alive[sl]) {
                    float v = s_bestv[sl];
                    long long fi = (long long)s_bestr[sl] * Dd + s_mlist[sl];
                    if (v > bv || (v == bv && fi < bidx)) { bv = v; bidx = fi; bslot = sl; }
                }
            }
            int col = s_mlist[bslot];
            s_col = col; s_colseq[t] = col;
            s_alive[bslot] = 0; s_miss[col] = 0.0f;
            g_colseq[t] = col;
            g_bsel[t]   = bvec[col];
        }
        __syncthreads();
        // Newly-known row s_col may improve remaining columns' maxima.
        if (tid < Kk && s_alive[tid]) {
            const int j = s_mlist[tid];
            float v = corr[s_col * Dd + j];
            if (v > s_bestv[tid] || (v == s_bestv[tid] && s_col < s_bestr[tid])) {
                s_bestv[tid] = v; s_bestr[tid] = s_col;
            }
        }
        __syncthreads();
    }
    // wseqT[t][s] = w[col_s, col_t] for s < t, else 0  (transposed: contiguous in s)
    for (int idx = tid; idx < Kk * Kk; idx += 256) {
        int trow = idx / Kk, scol = idx % Kk;
        g_wseqT[idx] = (scol < trow) ? w[s_colseq[scol] * Dd + s_colseq[trow]] : 0.0f;
    }
    // wmT[n][d] = w[d, col_n] * known0[d]  (N-major for float2 B-frag loads)
    for (int idx = tid; idx < Kk * Dd; idx += 256) {
        int n = idx & (Kk - 1);
        int d = idx >> 7;
        g_wmT[n * Dd + d] = w[d * Dd + s_colseq[n]] * s_known0[d];
    }
}

// ---------------------------------------------------------------------------
// Kernel 2: base = vals @ wmT^T via V_WMMA_F32_16X16X4_F32.
// One wave computes a 16(M)x64(N) tile with 4 accumulators: each A fragment
// is reused across 4 WMMAs (1.25 loads/WMMA instead of 2), B sub-tile
// pointers differ only by immediate offsets (16*Dd floats = 32 KB).
// ---------------------------------------------------------------------------
__global__ void __launch_bounds__(256) gemm_kernel(
    const float* __restrict__ in0, const float* __restrict__ wmT,
    float* __restrict__ base)
{
    const int lane = threadIdx.x & 31;
    const int wave = threadIdx.x >> 5;
    const int id   = blockIdx.x * 8 + wave;   // 512 waves total
    const int m0   = (id >> 1) * 16;
    const int n0   = (id & 1) * 64;
    const int half = lane >> 4;               // 0: K=k,k+1   1: K=k+2,k+3
    const int l15  = lane & 15;

    const float* aRow = in0 + (size_t)(m0 + l15) * (2 * Dd);  // vals row, stride 2D
    const float* bCol = wmT + (size_t)(n0 + l15) * Dd;        // N-major column

    v8f acc0 = {}, acc1 = {}, acc2 = {}, acc3 = {};
#pragma unroll 2
    for (int k = 0; k < Dd; k += 4) {
        const int kb = k + half * 2;
        v2f a  = *(const v2f*)(aRow + kb);
        v2f b0 = *(const v2f*)(bCol + kb);
        v2f b1 = *(const v2f*)(bCol + 16 * Dd + kb);
        v2f b2 = *(const v2f*)(bCol + 32 * Dd + kb);
        v2f b3 = *(const v2f*)(bCol + 48 * Dd + kb);
        acc0 = __builtin_amdgcn_wmma_f32_16x16x4_f32(false, a, false, b0, (short)0, acc0, false, false);
        acc1 = __builtin_amdgcn_wmma_f32_16x16x4_f32(false, a, false, b1, (short)0, acc1, false, false);
        acc2 = __builtin_amdgcn_wmma_f32_16x16x4_f32(false, a, false, b2, (short)0, acc2, false, false);
        acc3 = __builtin_amdgcn_wmma_f32_16x16x4_f32(false, a, false, b3, (short)0, acc3, false, false);
    }
    // C/D layout: VGPR v -> M = m0 + v + half*8, N = n0 + (lane&15) + 16*j
    float* op = base + (size_t)(m0 + half * 8) * Kk + (n0 + l15);
#pragma unroll
    for (int v = 0; v < 8; ++v) {
        op[(size_t)v * Kk + 0]  = acc0[v];
        op[(size_t)v * Kk + 16] = acc1[v];
        op[(size_t)v * Kk + 32] = acc2[v];
        op[(size_t)v * Kk + 48] = acc3[v];
    }
}

// ---------------------------------------------------------------------------
// Kernel 3: out = vals (float4-vectorized copy; K columns overwritten later).
// ---------------------------------------------------------------------------
__global__ void __launch_bounds__(256) copy_kernel(
    const float* __restrict__ in0, float* __restrict__ out)
{
    const int i  = blockIdx.x * 256 + threadIdx.x;   // over B*D/4 elements
    const int m  = i >> 7;                           // 512/4 = 128 float4 per row
    const int d4 = i & 127;
    ((float4*)out)[(size_t)m * 128 + d4] =
        ((const float4*)in0)[(size_t)m * 256 + d4];  // row stride 2D floats
}

// ---------------------------------------------------------------------------
// Kernel 4: forward substitution per batch row:
//   n_t = base[m,t] + b[col_t] + sum_{s<t} n_s * wseqT[t,s];  out[m,col_t]=n_t
// 4 independent partial accumulators break the FMA dependence chain; the
// transposed wseqT makes weight reads contiguous in s (merge to ds_load_b128);
// s_n[s][tid] history stays bank-conflict-free (bank = tid & 63).
// ---------------------------------------------------------------------------
__global__ void __launch_bounds__(128) tri_kernel(
    const float* __restrict__ base, const float* __restrict__ g_wseqT,
    const int* __restrict__ g_colseq, const float* __restrict__ g_bsel,
    float* __restrict__ out)
{
    __shared__ float s_wt[Kk * Kk];     // [t][s], 64 KB
    __shared__ float s_n[Kk * 128];     // [s][tid], 64 KB
    __shared__ int   s_cols[Kk];
    __shared__ float s_b[Kk];

    const int tid = threadIdx.x;
    for (int i = tid; i < Kk * Kk; i += 128) s_wt[i] = g_wseqT[i];
    if (tid < Kk) { s_cols[tid] = g_colseq[tid]; s_b[tid] = g_bsel[tid]; }
    __syncthreads();

    const int m = blockIdx.x * 128 + tid;
    const float* brow = base + (size_t)m * Kk;
    float* orow = out + (size_t)m * Dd;

    for (int t = 0; t < Kk; ++t) {
        const float* wt = &s_wt[t * Kk];
        float a0 = brow[t] + s_b[t];
        float a1 = 0.0f, a2 = 0.0f, a3 = 0.0f;
        const int s4 = t & ~3;
        for (int s = 0; s < s4; s += 4) {
            a0 += s_n[(s + 0) * 128 + tid] * wt[s + 0];
            a1 += s_n[(s + 1) * 128 + tid] * wt[s + 1];
            a2 += s_n[(s + 2) * 128 + tid] * wt[s + 2];
            a3 += s_n[(s + 3) * 128 + tid] * wt[s + 3];
        }
        for (int s = s4; s < t; ++s)
            a0 += s_n[s * 128 + tid] * wt[s];
        const float acc = (a0 + a1) + (a2 + a3);
        s_n[t * 128 + tid] = acc;
        orow[s_cols[t]] = acc;
    }
}

extern "C" void kernel_launch(void* const* d_in, const int* in_sizes, int n_in,
                              void* d_out, int out_size, void* d_ws, size_t ws_size,
                              hipStream_t stream) {
    const float* in0  = (const float*)d_in[0];  // inputs (B,2,D)
    const float* w    = (const float*)d_in[1];  // (D,D)
    const float* bv   = (const float*)d_in[2];  // (D,)
    const float* corr = (const float*)d_in[3];  // (D,D)
    float* out = (float*)d_out;

    char* ws = (char*)d_ws;
    int*   g_colseq = (int*)(ws);                                    // 512 B
    float* g_bsel   = (float*)(ws + 512);                            // 512 B
    float* g_wseqT  = (float*)(ws + 1024);                           // 64 KB
    float* g_wmT    = (float*)(ws + 1024 + 65536);                   // 256 KB
    float* g_base   = (float*)(ws + 1024 + 65536 + 262144);          // 2 MB

    seq_kernel<<<1, 256, 0, stream>>>(in0, w, bv, corr, g_colseq, g_bsel, g_wseqT, g_wmT);
    gemm_kernel<<<(Bb / 16) * (Kk / 64) / 8, 256, 0, stream>>>(in0, g_wmT, g_base);
    copy_kernel<<<(Bb * Dd / 4) / 256, 256, 0, stream>>>(in0, out);
    tri_kernel<<<Bb / 128, 128, 0, stream>>>(g_base, g_wseqT, g_colseq, g_bsel, out);
}